// RANSAC_35648228556820
// MI455X (gfx1250) — compile-verified
//
#include <hip/hip_runtime.h>
#include <math.h>

// ---------------------------------------------------------------- constants
#define N_PTS      10000
#define RBATCH     2048
#define MAX_ITER   10
#define TOTAL_HYP  (RBATCH * MAX_ITER)      // 20480
#define INL_TH     2.0f
#define REPS       1e-8f
#define N_TILES    (N_PTS / 16)             // 625
#define WAVES_PER_BLOCK 8

typedef __attribute__((ext_vector_type(2))) float v2f;
typedef __attribute__((ext_vector_type(8))) float v8f;

// Fast hardware reciprocal (v_rcp_f32, ~1 ulp) — accurate enough for
// threshold classification; avoids the IEEE div_scale/div_fmas chain.
__device__ __forceinline__ float fast_rcp(float x) {
    return __builtin_amdgcn_rcpf(x);
}

// ---------------------------------------------------------------- helpers
__device__ __forceinline__ unsigned pcg_hash(unsigned x) {
    x ^= x >> 16; x *= 0x7feb352du;
    x ^= x >> 15; x *= 0x846ca68bu;
    x ^= x >> 16;
    return x;
}

__device__ __forceinline__ void mat3_mul(float* __restrict__ c,
                                         const float* __restrict__ a,
                                         const float* __restrict__ b) {
#pragma unroll
    for (int i = 0; i < 3; ++i)
#pragma unroll
        for (int j = 0; j < 3; ++j)
            c[i*3+j] = a[i*3+0]*b[0*3+j] + a[i*3+1]*b[1*3+j] + a[i*3+2]*b[2*3+j];
}

// ---------------------------------------------------------------- 0) zero counters
__global__ void ransac_zero_kernel(int* __restrict__ counts) {
    int t = blockIdx.x * blockDim.x + threadIdx.x;
    if (t < TOTAL_HYP) counts[t] = 0;
}

// ---------------------------------------------------------------- 1) DLT per hypothesis
__global__ void __launch_bounds__(128)
ransac_dlt_kernel(const float2* __restrict__ kp1, const float2* __restrict__ kp2,
                  float* __restrict__ Hws, float* __restrict__ Hiws,
                  int* __restrict__ validws) {
    const int hyp = blockIdx.x * blockDim.x + threadIdx.x;
    if (hyp >= TOTAL_HYP) return;

    // ---- deterministic 4-distinct-index sampling
    int idx[4];
#pragma unroll
    for (int j = 0; j < 4; ++j) {
        unsigned v = pcg_hash((unsigned)hyp * 0x9E3779B9u + (unsigned)j * 0x85EBCA6Bu + 0x1234567u)
                     % (unsigned)N_PTS;
        bool dup = true;
        while (dup) {
            dup = false;
            for (int k = 0; k < j; ++k)
                if (idx[k] == (int)v) { v = (v + 1u) % (unsigned)N_PTS; dup = true; }
        }
        idx[j] = (int)v;
    }

    float2 p1[4], p2[4];
#pragma unroll
    for (int j = 0; j < 4; ++j) { p1[j] = kp1[idx[j]]; p2[j] = kp2[idx[j]]; }

    // ---- Hartley normalization for both point sets
    float mx1 = 0.f, my1 = 0.f, mx2 = 0.f, my2 = 0.f;
#pragma unroll
    for (int j = 0; j < 4; ++j) { mx1 += p1[j].x; my1 += p1[j].y; mx2 += p2[j].x; my2 += p2[j].y; }
    mx1 *= 0.25f; my1 *= 0.25f; mx2 *= 0.25f; my2 *= 0.25f;
    float d1 = 0.f, d2 = 0.f;
#pragma unroll
    for (int j = 0; j < 4; ++j) {
        float ax = p1[j].x - mx1, ay = p1[j].y - my1;
        float bx = p2[j].x - mx2, by = p2[j].y - my2;
        d1 += sqrtf(ax*ax + ay*ay);
        d2 += sqrtf(bx*bx + by*by);
    }
    d1 *= 0.25f; d2 *= 0.25f;
    const float s1 = 1.41421356237f * fast_rcp(d1 + REPS);
    const float s2 = 1.41421356237f * fast_rcp(d2 + REPS);

    // ---- accumulate AtA (9x9 symmetric) from 8 constraint rows
    float AtA[81];
#pragma unroll
    for (int i = 0; i < 81; ++i) AtA[i] = 0.f;
#pragma unroll
    for (int j = 0; j < 4; ++j) {
        const float x1 = (p1[j].x - mx1) * s1, y1 = (p1[j].y - my1) * s1;
        const float x2 = (p2[j].x - mx2) * s2, y2 = (p2[j].y - my2) * s2;
        const float ax[9] = { 0.f, 0.f, 0.f, -x1, -y1, -1.f,  y2*x1,  y2*y1,  y2 };
        const float ay[9] = { x1,  y1,  1.f,  0.f, 0.f,  0.f, -x2*x1, -x2*y1, -x2 };
#pragma unroll
        for (int a = 0; a < 9; ++a)
#pragma unroll
            for (int b = 0; b < 9; ++b)
                AtA[a*9+b] += ax[a]*ax[b] + ay[a]*ay[b];
    }

    // ---- shifted LU (no pivoting; AtA is SPD-ish) -> inverse iteration for null vector
    float M[81];
#pragma unroll
    for (int i = 0; i < 81; ++i) M[i] = AtA[i];
#pragma unroll
    for (int d = 0; d < 9; ++d) M[d*9+d] += 1e-6f;

#pragma unroll
    for (int k = 0; k < 9; ++k) {
        float piv = M[k*9+k];
        piv = (fabsf(piv) > 1e-20f) ? piv : 1e-20f;
        const float pinv = fast_rcp(piv);
#pragma unroll
        for (int i = k + 1; i < 9; ++i) {
            const float f = M[i*9+k] * pinv;
            M[i*9+k] = f;
#pragma unroll
            for (int c = k + 1; c < 9; ++c) M[i*9+c] -= f * M[k*9+c];
        }
    }

    float xv[9];
#pragma unroll
    for (int i = 0; i < 9; ++i) xv[i] = 0.33333333f;
#pragma unroll
    for (int it = 0; it < 6; ++it) {
        float y[9];
#pragma unroll
        for (int i = 0; i < 9; ++i) {
            float s = xv[i];
#pragma unroll
            for (int c = 0; c < 9; ++c) if (c < i) s -= M[i*9+c] * y[c];
            y[i] = s;
        }
        float z[9];
#pragma unroll
        for (int ii = 8; ii >= 0; --ii) {
            float s = y[ii];
#pragma unroll
            for (int c = 0; c < 9; ++c) if (c > ii) s -= M[ii*9+c] * z[c];
            float dg = M[ii*9+ii];
            dg = (fabsf(dg) > 1e-20f) ? dg : 1e-20f;
            z[ii] = s * fast_rcp(dg);
        }
        float nrm = 1e-30f;
#pragma unroll
        for (int i = 0; i < 9; ++i) nrm += z[i]*z[i];
        const float inv = rsqrtf(nrm);
#pragma unroll
        for (int i = 0; i < 9; ++i) xv[i] = z[i] * inv;
    }

    // ---- un-normalize: H = inv(T2) * Hn * T1, then /= H[2][2]
    const float is2 = fast_rcp(s2);
    const float T1[9]  = { s1, 0.f, -s1*mx1,   0.f, s1, -s1*my1,   0.f, 0.f, 1.f };
    const float T2i[9] = { is2, 0.f, mx2,   0.f, is2, my2,   0.f, 0.f, 1.f };
    float tmp[9], H[9];
    mat3_mul(tmp, xv, T1);
    mat3_mul(H, T2i, tmp);
    const float dn = fast_rcp(H[8] + REPS);
#pragma unroll
    for (int i = 0; i < 9; ++i) H[i] *= dn;

    // ---- adjugate inverse (scale-free use downstream)
    const float det = H[0]*(H[4]*H[8]-H[5]*H[7])
                    - H[1]*(H[3]*H[8]-H[5]*H[6])
                    + H[2]*(H[3]*H[7]-H[4]*H[6]);
    const float idet = fast_rcp(det + ((det >= 0.f) ? 1e-30f : -1e-30f));
    float Hi[9];
    Hi[0] = (H[4]*H[8]-H[5]*H[7])*idet;
    Hi[1] = (H[2]*H[7]-H[1]*H[8])*idet;
    Hi[2] = (H[1]*H[5]-H[2]*H[4])*idet;
    Hi[3] = (H[5]*H[6]-H[3]*H[8])*idet;
    Hi[4] = (H[0]*H[8]-H[2]*H[6])*idet;
    Hi[5] = (H[2]*H[3]-H[0]*H[5])*idet;
    Hi[6] = (H[3]*H[7]-H[4]*H[6])*idet;
    Hi[7] = (H[1]*H[6]-H[0]*H[7])*idet;
    Hi[8] = (H[0]*H[4]-H[1]*H[3])*idet;

    const float dmin = fminf(fabsf(H[0]), fminf(fabsf(H[4]), fabsf(H[8])));

    float* __restrict__ ho = Hws  + (size_t)hyp * 9;
    float* __restrict__ io = Hiws + (size_t)hyp * 9;
#pragma unroll
    for (int i = 0; i < 9; ++i) { ho[i] = H[i]; io[i] = Hi[i]; }
    validws[hyp] = (dmin > 1e-6f) ? 1 : 0;
}

// ---------------------------------------------------------------- 2) WMMA inlier counting
// Tile: M = 16 hypotheses, N = 16 points, K = 4 (x, 1, y, 0).
// Three WMMAs per direction produce Dx/Dy/Dw with identical lane layouts,
// so the projective divide + error compare is purely per-lane. The rcp is a
// TRANS op, so the epilogue co-executes with the WMMA (XDL) pipe.
__global__ void __launch_bounds__(256)
ransac_count_kernel(const float2* __restrict__ kp1, const float2* __restrict__ kp2,
                    const float* __restrict__ Hws, const float* __restrict__ Hiws,
                    int* __restrict__ counts) {
    const int lane = threadIdx.x & 31;
    const int wave = threadIdx.x >> 5;
    const int half = lane >> 4;        // 0: K=0..1 ; 1: K=2..3
    const int m    = lane & 15;
    const int hypBase = blockIdx.x * 16;

    const float* __restrict__ Hp  = Hws  + (size_t)(hypBase + m) * 9;
    const float* __restrict__ Hip = Hiws + (size_t)(hypBase + m) * 9;

    // A operands: row r of each hypothesis. Lanes 0-15 carry (h_r0, h_r1),
    // lanes 16-31 carry (h_r2, 0) per the 32-bit 16x4 A layout.
    v2f Af[3], Ab[3];
#pragma unroll
    for (int r = 0; r < 3; ++r) {
        Af[r][0] = half ? Hp[r*3+2]  : Hp[r*3+0];
        Af[r][1] = half ? 0.0f       : Hp[r*3+1];
        Ab[r][0] = half ? Hip[r*3+2] : Hip[r*3+0];
        Ab[r][1] = half ? 0.0f       : Hip[r*3+1];
    }

    int cnt[8];
#pragma unroll
    for (int r = 0; r < 8; ++r) cnt[r] = 0;

    const v8f cz = {0.f,0.f,0.f,0.f,0.f,0.f,0.f,0.f};

    for (int pt = wave; pt < N_TILES; pt += WAVES_PER_BLOCK) {
        const int n = pt * 16 + m;
        const float2 p1 = kp1[n];
        const float2 p2 = kp2[n];
        if (pt + WAVES_PER_BLOCK < N_TILES) {          // wave-uniform branch
            __builtin_prefetch(&kp1[(pt + WAVES_PER_BLOCK) * 16 + m], 0, 1);
            __builtin_prefetch(&kp2[(pt + WAVES_PER_BLOCK) * 16 + m], 0, 1);
        }

        // B operands: 4x16 (K rows x N points). VGPR0 = rows K0 (x) / K2 (y),
        // VGPR1 = rows K1 (1) / K3 (0).
        v2f B1, B2;
        B1[0] = half ? p1.y : p1.x;  B1[1] = half ? 0.0f : 1.0f;
        B2[0] = half ? p2.y : p2.x;  B2[1] = half ? 0.0f : 1.0f;

        v8f Dx1 = __builtin_amdgcn_wmma_f32_16x16x4_f32(false, Af[0], false, B1, (short)0, cz, false, false);
        v8f Dy1 = __builtin_amdgcn_wmma_f32_16x16x4_f32(false, Af[1], false, B1, (short)0, cz, false, false);
        v8f Dw1 = __builtin_amdgcn_wmma_f32_16x16x4_f32(false, Af[2], false, B1, (short)0, cz, false, false);
        v8f Dx2 = __builtin_amdgcn_wmma_f32_16x16x4_f32(false, Ab[0], false, B2, (short)0, cz, false, false);
        v8f Dy2 = __builtin_amdgcn_wmma_f32_16x16x4_f32(false, Ab[1], false, B2, (short)0, cz, false, false);
        v8f Dw2 = __builtin_amdgcn_wmma_f32_16x16x4_f32(false, Ab[2], false, B2, (short)0, cz, false, false);

        // Element (VGPR r, this lane) is (hyp = hypBase + r + 8*half, point = n).
#pragma unroll
        for (int r = 0; r < 8; ++r) {
            const float w1 = Dw1[r];
            const float sc1 = (fabsf(w1) > REPS) ? fast_rcp(w1) : 1.0f;
            const float du1 = Dx1[r]*sc1 - p2.x;
            const float dv1 = Dy1[r]*sc1 - p2.y;
            const float w2 = Dw2[r];
            const float sc2 = (fabsf(w2) > REPS) ? fast_rcp(w2) : 1.0f;
            const float du2 = Dx2[r]*sc2 - p1.x;
            const float dv2 = Dy2[r]*sc2 - p1.y;
            const float err = du1*du1 + dv1*dv1 + du2*du2 + dv2*dv2;
            cnt[r] += (err <= INL_TH) ? 1 : 0;
        }
    }

    // Reduce over the 16 point-lanes of each half (xor masks stay in-half).
#pragma unroll
    for (int r = 0; r < 8; ++r) {
        int c = cnt[r];
        c += __shfl_xor(c, 1, warpSize);
        c += __shfl_xor(c, 2, warpSize);
        c += __shfl_xor(c, 4, warpSize);
        c += __shfl_xor(c, 8, warpSize);
        if (m == 0) atomicAdd(&counts[hypBase + r + 8 * half], c);
    }
}

// ---------------------------------------------------------------- 3) sequential-RANSAC selection
__global__ void __launch_bounds__(256)
ransac_select_kernel(const int* __restrict__ counts, const int* __restrict__ validws,
                     int* __restrict__ winner) {
    __shared__ float ss[256];
    __shared__ int   si[256];
    const int t = threadIdx.x;

    float bestScore = 1.0f;   // matches reference init
    int   bestIdx   = -1;

    for (int iter = 0; iter < MAX_ITER; ++iter) {
        float ls = -2.0f; int li = 0x7fffffff;
        for (int b = t; b < RBATCH; b += 256) {
            const int g = iter * RBATCH + b;
            const float sc = validws[g] ? (float)counts[g] : -1.0f;
            if (sc > ls || (sc == ls && g < li)) { ls = sc; li = g; }
        }
        ss[t] = ls; si[t] = li;
        __syncthreads();
        for (int off = 128; off > 0; off >>= 1) {
            if (t < off) {
                if (ss[t+off] > ss[t] || (ss[t+off] == ss[t] && si[t+off] < si[t])) {
                    ss[t] = ss[t+off]; si[t] = si[t+off];
                }
            }
            __syncthreads();
        }
        if (t == 0) {
            if (ss[0] > bestScore) { bestScore = ss[0]; bestIdx = si[0]; }
        }
        __syncthreads();
    }
    if (t == 0) winner[0] = bestIdx;
}

// ---------------------------------------------------------------- 4) emit best H + inlier mask
__global__ void __launch_bounds__(256)
ransac_output_kernel(const float2* __restrict__ kp1, const float2* __restrict__ kp2,
                     const float* __restrict__ Hws, const float* __restrict__ Hiws,
                     const int* __restrict__ winner, float* __restrict__ out) {
    const int tid = blockIdx.x * blockDim.x + threadIdx.x;
    const int w = winner[0];

    float H[9], Hi[9];
#pragma unroll
    for (int i = 0; i < 9; ++i) {
        H[i]  = (w >= 0) ? Hws[(size_t)w * 9 + i]  : ((i % 4 == 0) ? 1.0f : 0.0f);
        Hi[i] = (w >= 0) ? Hiws[(size_t)w * 9 + i] : ((i % 4 == 0) ? 1.0f : 0.0f);
    }
    if (tid < 9) out[tid] = H[tid];

    if (tid < N_PTS) {
        float mask = 0.0f;
        if (w >= 0) {
            const float2 a = kp1[tid], b = kp2[tid];
            float X = H[0]*a.x + H[1]*a.y + H[2];
            float Y = H[3]*a.x + H[4]*a.y + H[5];
            float Z = H[6]*a.x + H[7]*a.y + H[8];
            float s = (fabsf(Z) > REPS) ? fast_rcp(Z) : 1.0f;
            float du = X*s - b.x, dv = Y*s - b.y;
            float e1 = du*du + dv*dv;
            X = Hi[0]*b.x + Hi[1]*b.y + Hi[2];
            Y = Hi[3]*b.x + Hi[4]*b.y + Hi[5];
            Z = Hi[6]*b.x + Hi[7]*b.y + Hi[8];
            s = (fabsf(Z) > REPS) ? fast_rcp(Z) : 1.0f;
            du = X*s - a.x; dv = Y*s - a.y;
            const float e2 = du*du + dv*dv;
            mask = ((e1 + e2) <= INL_TH) ? 1.0f : 0.0f;
        }
        out[9 + tid] = mask;
    }
}

// ---------------------------------------------------------------- launch
extern "C" void kernel_launch(void* const* d_in, const int* in_sizes, int n_in,
                              void* d_out, int out_size, void* d_ws, size_t ws_size,
                              hipStream_t stream) {
    (void)in_sizes; (void)n_in; (void)out_size; (void)ws_size;
    const float2* kp1 = (const float2*)d_in[0];
    const float2* kp2 = (const float2*)d_in[1];
    float* out = (float*)d_out;

    char* ws = (char*)d_ws;
    const size_t offH  = 0;
    const size_t offHi = offH  + (size_t)TOTAL_HYP * 9 * sizeof(float);
    const size_t offV  = offHi + (size_t)TOTAL_HYP * 9 * sizeof(float);
    const size_t offC  = offV  + (size_t)TOTAL_HYP * sizeof(int);
    const size_t offW  = offC  + (size_t)TOTAL_HYP * sizeof(int);

    float* Hws   = (float*)(ws + offH);
    float* Hiws  = (float*)(ws + offHi);
    int*   valid = (int*)  (ws + offV);
    int*   cnts  = (int*)  (ws + offC);
    int*   win   = (int*)  (ws + offW);

    ransac_zero_kernel<<<(TOTAL_HYP + 255) / 256, 256, 0, stream>>>(cnts);
    ransac_dlt_kernel<<<TOTAL_HYP / 128, 128, 0, stream>>>(kp1, kp2, Hws, Hiws, valid);
    ransac_count_kernel<<<TOTAL_HYP / 16, 256, 0, stream>>>(kp1, kp2, Hws, Hiws, cnts);
    ransac_select_kernel<<<1, 256, 0, stream>>>(cnts, valid, win);
    ransac_output_kernel<<<(N_PTS + 255) / 256, 256, 0, stream>>>(kp1, kp2, Hws, Hiws, win, out);
}